// MyGaussianPDF_1391569403988
// MI455X (gfx1250) — compile-verified
//
#include <hip/hip_runtime.h>

// Gaussian PDF over N 2D points: out[n] = exp(-0.27 * ((x0-mu0)^2 + (x1-mu1)^2)).
// Streaming map, 12 B/point -> HBM-bound (~17 us floor at 23.3 TB/s on MI455X).
// CDNA5 path: wave-private depth-3 async-to-LDS pipeline (ASYNCcnt, no barriers),
// b128 loads, ds_load_b128 readback, v_exp_f32, non-temporal b64 stores.
// No WMMA: ~0.6 flop/byte is orders of magnitude below any matrix-op crossover.

typedef __attribute__((ext_vector_type(4))) float v4f;
typedef __attribute__((ext_vector_type(2))) float v2f;

#define TPB    256u   // 8 waves per block
#define BLOCKS 4096u  // compile-time grid stride: no u64 division in the kernel
#define STAGES 4u     // LDS ring buffers
#define DEPTH  3u     // async tiles in flight per wave (one stage of slack)

__device__ __forceinline__ float fast_exp2(float x) {
#if __has_builtin(__builtin_amdgcn_exp2f)
  return __builtin_amdgcn_exp2f(x);   // v_exp_f32
#else
  return exp2f(x);
#endif
}

__device__ __forceinline__ void issue_async_tile(const v4f* src, const v4f* lds_slot) {
  // GLOBAL_LOAD_ASYNC_TO_LDS_B128: each lane copies 16B from global memory to the LDS address in vdst.
  // Low 32 bits of a generic pointer to __shared__ are the LDS byte offset.
  unsigned lds_off = (unsigned)(unsigned long long)lds_slot;
  asm volatile("global_load_async_to_lds_b128 %0, %1, off"
               :: "v"(lds_off), "v"(src) : "memory");
}

__global__ __launch_bounds__(TPB) void gauss_pdf_async_kernel(
    const v4f* __restrict__ x4, const float* __restrict__ xf,
    const float* __restrict__ mu,
    v2f* __restrict__ out2, float* __restrict__ out1,
    unsigned n4, unsigned nFullTiles, unsigned nPoints)
{
  __shared__ v4f buf[STAGES][TPB];   // 16 KB of the 320 KB/WGP
  const unsigned tid = threadIdx.x;
  const float mu0 = mu[0];
  const float mu1 = mu[1];
  // exp(-0.27*q) == exp2(c2*q), c2 = -0.27 * log2(e)
  const float c2 = -0.38952766104020514f;
  const unsigned first = blockIdx.x;

  // ---- prologue: fill the pipeline (block-uniform control => EXEC all-ones)
#pragma unroll
  for (unsigned s = 0; s < DEPTH; ++s) {
    const unsigned t = first + s * BLOCKS;
    if (t < nFullTiles)
      issue_async_tile(x4 + (size_t)t * TPB + tid, &buf[s][tid]);
  }

  // ---- steady state: wait oldest, read own LDS slot, compute, refill, store
  unsigned i = 0;
  for (unsigned t = first; t < nFullTiles; t += BLOCKS, ++i) {
    if (t + 2u * BLOCKS < nFullTiles) {
      // >=3 tiles remain -> 3 in flight; leaving 2 outstanding completes the
      // oldest (async loads retire in order within a wave).
      asm volatile("s_wait_asynccnt 2" ::: "memory");
    } else {
      asm volatile("s_wait_asynccnt 0" ::: "memory");   // drain tail
    }
    const unsigned s = i & (STAGES - 1u);
    v4f v = buf[s][tid];                                // ds_load_b128, lane-private

    float d0 = v.x - mu0, e0 = v.y - mu1;
    float d1 = v.z - mu0, e1 = v.w - mu1;
    float q0 = __builtin_fmaf(e0, e0, d0 * d0);
    float q1 = __builtin_fmaf(e1, e1, d1 * d1);
    v2f r;
    r.x = fast_exp2(c2 * q0);
    r.y = fast_exp2(c2 * q1);

    // Refill stage (i+DEPTH)&3 == the stage consumed LAST iteration: one full
    // iteration of slack between its ds_load and the async LDS overwrite.
    const unsigned tn = t + DEPTH * BLOCKS;
    if (tn < nFullTiles)
      issue_async_tile(x4 + (size_t)tn * TPB + tid,
                       &buf[(i + DEPTH) & (STAGES - 1u)][tid]);

    __builtin_nontemporal_store(r, out2 + (size_t)t * TPB + tid); // never re-read
  }

  // ---- remainder float4s not covered by full tiles (empty when N = 2^25)
  const unsigned gtid = first * TPB + tid;
  for (unsigned idx = nFullTiles * TPB + gtid; idx < n4; idx += BLOCKS * TPB) {
    v4f v = __builtin_nontemporal_load(x4 + idx);
    float d0 = v.x - mu0, e0 = v.y - mu1;
    float d1 = v.z - mu0, e1 = v.w - mu1;
    float q0 = __builtin_fmaf(e0, e0, d0 * d0);
    float q1 = __builtin_fmaf(e1, e1, d1 * d1);
    v2f r;
    r.x = fast_exp2(c2 * q0);
    r.y = fast_exp2(c2 * q1);
    __builtin_nontemporal_store(r, out2 + idx);
  }

  // ---- odd final point (N odd only)
  if ((nPoints & 1u) && gtid == 0) {
    const unsigned p = nPoints - 1u;
    float d0 = xf[2u * p + 0u] - mu0;
    float e0 = xf[2u * p + 1u] - mu1;
    float q0 = __builtin_fmaf(e0, e0, d0 * d0);
    out1[p] = fast_exp2(c2 * q0);
  }
}

extern "C" void kernel_launch(void* const* d_in, const int* in_sizes, int n_in,
                              void* d_out, int out_size, void* d_ws, size_t ws_size,
                              hipStream_t stream) {
  (void)n_in; (void)d_ws; (void)ws_size;
  const v4f*   x4 = (const v4f*)d_in[0];
  const float* xf = (const float*)d_in[0];
  const float* mu = (const float*)d_in[1];

  const unsigned nPoints    = (unsigned)out_size;              // N
  const unsigned n4         = (unsigned)in_sizes[0] / 4u;      // float4 count
  const unsigned nFullTiles = n4 / TPB;

  // 4096 blocks x 8 waves = 32K waves; 3 async b128 tiles in flight per wave
  // ~= 48 MB outstanding reads, above the ~23 MB bandwidth-delay product.
  gauss_pdf_async_kernel<<<BLOCKS, TPB, 0, stream>>>(
      x4, xf, mu, (v2f*)d_out, (float*)d_out, n4, nFullTiles, nPoints);
}